// GATLayer_48550310314658
// MI455X (gfx1250) — compile-verified
//
#include <hip/hip_runtime.h>
#include <hip/hip_bf16.h>
#include <stdint.h>

// GAT layer for MI455X (gfx1250), fp32 end-to-end.
//  N=4096 nodes, C_IN=C_OUT=256, H=4 heads, CC=64, E=131072 edges.
//
// Pipeline:
//  k0: zero 2MB adjacency bitmask (ws)          -- dedup semantics == dense adj
//  k1: scatter edges into bitmask (atomicOr)
//  k2: f = X@W + b via V_WMMA_F32_16X16X4_F32   (wave32 WMMA, fp32 exact)
//  k3: e_src/e_dst per (node, head)             (LDS tree reduce)
//  k4: per-node masked softmax + aggregation    (bit-scan row, three-pass softmax)

#define GAT_N      4096
#define GAT_C      256
#define GAT_H      4
#define GAT_CC     64
#define GAT_ALPHA  0.2f
#define MASK_WORDS (GAT_N * GAT_N / 32)   // 524288 u32 = 2MB

typedef float v2f __attribute__((ext_vector_type(2)));
typedef float v8f __attribute__((ext_vector_type(8)));

__device__ __forceinline__ float gat_lrelu(float x) {
    return x > 0.0f ? x : GAT_ALPHA * x;
}

// ---------------------------------------------------------------- k0: zero mask
__global__ __launch_bounds__(256) void gat_zero_mask(uint32_t* mask) {
    int idx = blockIdx.x * 256 + threadIdx.x;
    if (idx < MASK_WORDS) mask[idx] = 0u;
}

// ---------------------------------------------------------------- k1: scatter
__global__ __launch_bounds__(256) void gat_scatter_edges(const int* __restrict__ edges,
                                                         uint32_t* __restrict__ mask,
                                                         int E) {
    int e = blockIdx.x * 256 + threadIdx.x;
    if (e >= E) return;
    int i = edges[2 * e + 0];
    int j = edges[2 * e + 1];
    unsigned long long pos = (unsigned long long)i * GAT_N + (unsigned)j;
    atomicOr(&mask[pos >> 5], 1u << (pos & 31u));
}

// ---------------------------------------------------------------- k2: WMMA GEMM
// f[M=4096, N=256] = X[4096,256] @ W[256,256] + b
// grid = (32, 16): block covers 128 rows x 16 cols; 8 waves stacked in M.
__global__ __launch_bounds__(256) void gat_gemm_wmma(const float* __restrict__ X,
                                                     const float* __restrict__ W,
                                                     const float* __restrict__ bias,
                                                     float* __restrict__ f) {
    __shared__ float Bs[GAT_C * 16];       // W tile: all K x 16 cols (16KB)

    const int t       = threadIdx.x;
    const int ncol0   = blockIdx.y * 16;

    // Stage W tile in LDS (shared by all 8 waves).
    for (int idx = t; idx < GAT_C * 16; idx += 256) {
        int k = idx >> 4, n = idx & 15;
        Bs[idx] = W[k * GAT_C + ncol0 + n];
    }
    __syncthreads();

    const int wave = t >> 5;
    const int lane = t & 31;
    const int half = lane >> 4;            // 0: lanes 0-15, 1: lanes 16-31
    const int l16  = lane & 15;

    const int row = blockIdx.x * 128 + wave * 16 + l16;   // A row of this lane
    const float* __restrict__ Arow = X + (size_t)row * GAT_C;

    v8f acc = {};
    #pragma unroll 8
    for (int k = 0; k < GAT_C; k += 4) {
        // A 16x4 frag: lane holds A[row, k + 2*half + {0,1}]
        const int ka = k + 2 * half;
        v2f afrag = { Arow[ka], Arow[ka + 1] };
        // B 4x16 frag: vgpr v holds B[k + v + 2*half, ncol0 + l16]
        v2f bfrag = { Bs[(k + 2 * half)     * 16 + l16],
                      Bs[(k + 2 * half + 1) * 16 + l16] };
        acc = __builtin_amdgcn_wmma_f32_16x16x4_f32(
            /*neg_a=*/false, afrag, /*neg_b=*/false, bfrag,
            /*c_mod=*/(short)0, acc, /*reuse_a=*/false, /*reuse_b=*/false);
    }

    // C/D layout: vgpr r, lane -> M = r + 8*half, N = l16
    const float bcol  = bias[ncol0 + l16];
    const int   mBase = blockIdx.x * 128 + wave * 16 + 8 * half;
    #pragma unroll
    for (int r = 0; r < 8; ++r) {
        f[(size_t)(mBase + r) * GAT_C + ncol0 + l16] = acc[r] + bcol;
    }
}

// ---------------------------------------------------------------- k3: head logits
// e_src[i,h] = f[i,h,:] . a[h,:CC] ; e_dst[i,h] = f[i,h,:] . a[h,CC:]
__global__ __launch_bounds__(256) void gat_head_logits(const float* __restrict__ f,
                                                       const float* __restrict__ a,
                                                       float* __restrict__ esrc,
                                                       float* __restrict__ edst) {
    __shared__ float rs[256];
    __shared__ float rd[256];
    const int i = blockIdx.x;
    const int t = threadIdx.x;
    const int h = t >> 6, c = t & 63;

    float v = f[(size_t)i * GAT_C + t];
    rs[t] = v * a[h * (2 * GAT_CC) + c];
    rd[t] = v * a[h * (2 * GAT_CC) + GAT_CC + c];
    __syncthreads();
    for (int off = 32; off > 0; off >>= 1) {
        if (c < off) { rs[t] += rs[t + off]; rd[t] += rd[t + off]; }
        __syncthreads();
    }
    if (c == 0) {
        esrc[i * GAT_H + h] = rs[t];
        edst[i * GAT_H + h] = rd[t];
    }
}

// ---------------------------------------------------------------- k4: softmax+agg
// One block per node i. Scan 128 mask words, build neighbor list in LDS,
// per-head max -> sum(exp) -> weighted accumulation (thread = (h,c)).
__global__ __launch_bounds__(256) void gat_aggregate(const uint32_t* __restrict__ mask,
                                                     const float* __restrict__ f,
                                                     const float* __restrict__ esrc,
                                                     const float* __restrict__ edst,
                                                     float* __restrict__ out) {
    __shared__ uint16_t nbr[GAT_N];        // 8KB neighbor indices
    __shared__ float    red[256];
    __shared__ float    smax[GAT_H];
    __shared__ float    ssum[GAT_H];
    __shared__ int      cnt;

    const int i = blockIdx.x;
    const int t = threadIdx.x;
    if (t == 0) cnt = 0;
    __syncthreads();

    float si[GAT_H];
    #pragma unroll
    for (int h = 0; h < GAT_H; ++h) si[h] = esrc[i * GAT_H + h];

    // --- pass A: bit-scan row, build list, per-thread running max per head ---
    float lmax[GAT_H];
    #pragma unroll
    for (int h = 0; h < GAT_H; ++h) lmax[h] = -3.402823e38f;

    if (t < GAT_N / 32) {
        uint32_t w = mask[(size_t)i * (GAT_N / 32) + t];
        while (w) {
            int b = __ffs(w) - 1;
            w &= w - 1;
            int j = (t << 5) + b;
            int p = atomicAdd(&cnt, 1);
            nbr[p] = (uint16_t)j;
            #pragma unroll
            for (int h = 0; h < GAT_H; ++h) {
                float l = gat_lrelu(si[h] + edst[j * GAT_H + h]);
                lmax[h] = fmaxf(lmax[h], l);
            }
        }
    }
    __syncthreads();

    // reduce max per head
    for (int h = 0; h < GAT_H; ++h) {
        red[t] = lmax[h];
        __syncthreads();
        for (int off = 128; off > 0; off >>= 1) {
            if (t < off) red[t] = fmaxf(red[t], red[t + off]);
            __syncthreads();
        }
        if (t == 0) smax[h] = red[0];
        __syncthreads();
    }

    const int deg = cnt;

    // --- pass B: sum of exp per head ---
    float lsum[GAT_H] = {0.f, 0.f, 0.f, 0.f};
    for (int k = t; k < deg; k += 256) {
        int j = nbr[k];
        #pragma unroll
        for (int h = 0; h < GAT_H; ++h) {
            float l = gat_lrelu(si[h] + edst[j * GAT_H + h]);
            lsum[h] += expf(l - smax[h]);
        }
    }
    for (int h = 0; h < GAT_H; ++h) {
        red[t] = lsum[h];
        __syncthreads();
        for (int off = 128; off > 0; off >>= 1) {
            if (t < off) red[t] += red[t + off];
            __syncthreads();
        }
        if (t == 0) ssum[h] = red[0];
        __syncthreads();
    }

    // --- pass C: weighted aggregation; thread owns output channel (h,c) ---
    const int h = t >> 6;   // t = h*64 + c; f index j*256 + t
    float acc = 0.0f;
    if (deg > 0) {
        const float sih = si[h];
        const float mx  = smax[h];
        for (int k = 0; k < deg; ++k) {
            int j = nbr[k];
            float l = gat_lrelu(sih + edst[j * GAT_H + h]);
            acc += expf(l - mx) * f[(size_t)j * GAT_C + t];
        }
        out[(size_t)i * GAT_C + t] = acc / ssum[h];
    } else {
        // All logits == NEG_INF -> softmax is uniform over all N nodes.
        for (int j = 0; j < GAT_N; ++j) acc += f[(size_t)j * GAT_C + t];
        out[(size_t)i * GAT_C + t] = acc * (1.0f / (float)GAT_N);
    }
}

// ---------------------------------------------------------------- launch
extern "C" void kernel_launch(void* const* d_in, const int* in_sizes, int n_in,
                              void* d_out, int out_size, void* d_ws, size_t ws_size,
                              hipStream_t stream) {
    (void)n_in; (void)out_size; (void)ws_size;

    const float* X     = (const float*)d_in[0];       // [N, 256]
    const int*   edges = (const int*)d_in[1];         // [E, 2] (harness: integer -> int32)
    const float* W     = (const float*)d_in[2];       // [256, 256]
    const float* bias  = (const float*)d_in[3];       // [256]
    const float* a     = (const float*)d_in[4];       // [H, 2*CC]
    float*       out   = (float*)d_out;               // [1, N, 256]

    const int E = in_sizes[1] / 2;

    // workspace layout
    char* ws = (char*)d_ws;
    uint32_t* mask = (uint32_t*)(ws);                          // 2 MB
    float*    f    = (float*)(ws + (size_t)2 * 1024 * 1024);   // 4 MB
    float*    esrc = (float*)(ws + (size_t)6 * 1024 * 1024);   // 64 KB
    float*    edst = (float*)(ws + (size_t)6 * 1024 * 1024 + 64 * 1024); // 64 KB

    gat_zero_mask<<<(MASK_WORDS + 255) / 256, 256, 0, stream>>>(mask);
    gat_scatter_edges<<<(E + 255) / 256, 256, 0, stream>>>(edges, mask, E);

    dim3 ggrid(GAT_N / 128, GAT_C / 16);   // (32, 16)
    gat_gemm_wmma<<<ggrid, 256, 0, stream>>>(X, W, bias, f);

    gat_head_logits<<<GAT_N, 256, 0, stream>>>(f, a, esrc, edst);
    gat_aggregate<<<GAT_N, 256, 0, stream>>>(mask, f, esrc, edst, out);
}